// OptimalTransportDepthLoss_63178968924664
// MI455X (gfx1250) — compile-verified
//
#include <hip/hip_runtime.h>
#include <stdint.h>

#define D 256
#define ROWS 64          // rows per block == threads per block
#define NT 64
#define STR 257          // padded LDS row stride in floats (257 % 64 == 1)

// ---------------------------------------------------------------------------
// Kernel 1: one thread = one row. Async-stage 64 rows of P and Q into LDS,
// then per-thread: row sums + MSE, then fused cumsum + two-pointer merge EMD^2.
// ---------------------------------------------------------------------------
__global__ __launch_bounds__(NT)
void otdl_rows_kernel(const float* __restrict__ P, const float* __restrict__ Q,
                      float* __restrict__ blockSums) {
  __shared__ float ldsbuf[2 * ROWS * STR];   // 131584 bytes
  float* ldsP = ldsbuf;
  float* ldsQ = ldsbuf + ROWS * STR;

  const int tid = threadIdx.x;
  const size_t tileBase = (size_t)blockIdx.x * (ROWS * D);
  const float* gP = P + tileBase;
  const float* gQ = Q + tileBase;

  // Low 32 bits of the flat address of a __shared__ object == LDS byte offset
  // (amdgcn AS3->AS0 cast puts the LDS offset in the low dword).
  uint32_t lpBase = (uint32_t)(uintptr_t)ldsP;
  uint32_t lqBase = (uint32_t)(uintptr_t)ldsQ;

  // Phase 1: CDNA5 async global -> LDS copy (GVS mode: SGPR base + VGPR u32
  // byte offset), writing the row-padded [r][257] layout. Element
  // e = k*64 + tid  ->  r = e/256, c = e%256. Global reads coalesced (128B
  // per wave issue); LDS write banks (r*257 + c) % 64 are conflict-free.
  #pragma unroll 4
  for (int k = 0; k < D; ++k) {                 // 256 iters, 2 arrays
    int e = (k << 6) + tid;                     // dword index in tile
    int r = e >> 8;
    int c = e & (D - 1);
    uint32_t loff = (uint32_t)((r * STR + c) << 2);
    uint32_t goff = (uint32_t)e << 2;
    uint32_t lp = lpBase + loff;
    uint32_t lq = lqBase + loff;
    asm volatile("global_load_async_to_lds_b32 %0, %1, %2"
                 :: "v"(lp), "v"(goff), "s"(gP) : "memory");
    asm volatile("global_load_async_to_lds_b32 %0, %1, %2"
                 :: "v"(lq), "v"(goff), "s"(gQ) : "memory");
  }
  asm volatile("s_wait_asynccnt 0" ::: "memory");
  __syncthreads();

  // Phase 2: per-row sums and MSE. Bank = (tid + i) % 64: conflict-free.
  const float* rp = ldsP + tid * STR;
  const float* rq = ldsQ + tid * STR;
  float sp = 0.f, sq = 0.f, sd2 = 0.f;
  #pragma unroll 4
  for (int i = 0; i < D; ++i) {
    float a = rp[i], b = rq[i];
    sp += a; sq += b;
    float dd = a - b;
    sd2 = fmaf(dd, dd, sd2);
  }
  const float invSP = 1.0f / sp;
  const float invSQ = 1.0f / sq;

  // Phase 3: exact EMD^2 by merging the two normalized CDFs.
  // Invariants: i,j = #consumed; cp = cp[i], cq = cq[j] (next unconsumed
  // CDF values, built incrementally); pn/qn = raw next-next elements
  // (one-ahead prefetch). Segment (tprev, t] has quantile indices
  // min(i,D-1), min(j,D-1) — identical to searchsorted at midpoints.
  const float INF = __builtin_huge_valf();
  int i = 0, j = 0;
  float cp = rp[0] * invSP;
  float cq = rq[0] * invSQ;
  float pn = rp[1];
  float qn = rq[1];
  float tprev = 0.f, acc = 0.f;
  const int qsel = ROWS * STR;   // offset from rp-row to rq-row in ldsbuf

  for (int k = 0; k < 2 * D; ++k) {
    bool takep = (j >= D) || ((i < D) && (cp <= cq));
    float t  = takep ? cp : cq;
    float dt = t - tprev;
    tprev = t;
    int ip = i > (D - 1) ? (D - 1) : i;
    int iq = j > (D - 1) ? (D - 1) : j;
    float dd = (float)(ip - iq);
    acc = fmaf(dt * dd, dd, acc);

    int   adv = (takep ? i : j) + 1;
    float nv  = takep ? pn : qn;
    float sc  = takep ? invSP : invSQ;
    float cur = takep ? cp : cq;
    float upd = (adv < D) ? fmaf(nv, sc, cur) : INF;
    cp = takep ? upd : cp;
    cq = takep ? cq  : upd;
    i  = takep ? adv : i;
    j  = takep ? j   : adv;

    int nidx = adv + 1;
    nidx = nidx > (D - 1) ? (D - 1) : nidx;
    float ld = rp[(takep ? 0 : qsel) + nidx];   // one-ahead refill
    pn = takep ? ld : pn;
    qn = takep ? qn : ld;
  }

  float loss = sd2 * (1.0f / (float)D) + acc;   // mse + lambda*EMD2 (lambda=1)

  // Deterministic in-block tree reduction of 64 row losses.
  __syncthreads();                // everyone done reading lds before reuse
  ldsbuf[tid] = loss;
  __syncthreads();
  #pragma unroll
  for (int s = NT / 2; s > 0; s >>= 1) {
    if (tid < s) ldsbuf[tid] += ldsbuf[tid + s];
    __syncthreads();
  }
  if (tid == 0) blockSums[blockIdx.x] = ldsbuf[0];
}

// ---------------------------------------------------------------------------
// Kernel 2: deterministic final reduction of per-block sums -> mean loss.
// ---------------------------------------------------------------------------
__global__ __launch_bounds__(256)
void otdl_reduce_kernel(const float* __restrict__ bs, float* __restrict__ out,
                        int nblocks, float invN) {
  __shared__ float s[256];
  float v = 0.f;
  for (int idx = threadIdx.x; idx < nblocks; idx += 256) v += bs[idx];
  s[threadIdx.x] = v;
  __syncthreads();
  #pragma unroll
  for (int st = 128; st > 0; st >>= 1) {
    if (threadIdx.x < st) s[threadIdx.x] += s[threadIdx.x + st];
    __syncthreads();
  }
  if (threadIdx.x == 0) out[0] = s[0] * invN;
}

extern "C" void kernel_launch(void* const* d_in, const int* in_sizes, int n_in,
                              void* d_out, int out_size, void* d_ws, size_t ws_size,
                              hipStream_t stream) {
  const float* P = (const float*)d_in[0];
  const float* Q = (const float*)d_in[1];
  float* out = (float*)d_out;
  float* ws  = (float*)d_ws;

  const int total  = in_sizes[0];        // B * D
  const int nrows  = total / D;          // 65536
  const int nblk   = nrows / ROWS;       // 1024

  otdl_rows_kernel<<<nblk, NT, 0, stream>>>(P, Q, ws);
  otdl_reduce_kernel<<<1, 256, 0, stream>>>(ws, out, nblk, 1.0f / (float)nrows);
}